// Gate_9517647528205
// MI455X (gfx1250) — compile-verified
//
#include <hip/hip_runtime.h>
#include <hip/hip_bf16.h>

typedef __attribute__((ext_vector_type(2))) float v2f;
typedef __attribute__((ext_vector_type(8))) float v8f;

#define H_DIM   2048
#define N_EXP   64
#define TOPK    8
#define TOK_PER_BLOCK 128   // 8 waves x 16 tokens
#define LDS_PITCH 65        // 64 + 1 pad -> conflict-free row & column access

// ---------------------------------------------------------------------------
// Fused: logits GEMM (WMMA f32 16x16x4) + softmax + top-8 + per-block aux partials
// ---------------------------------------------------------------------------
__global__ __launch_bounds__(256) void gate_main_kernel(
    const float* __restrict__ x,      // [N, 2048]
    const float* __restrict__ wgt,    // [64, 2048]
    const float* __restrict__ bias,   // [64]
    int*  __restrict__ out_idx,       // [N, 8]
    float* __restrict__ out_w,        // [N, 8]
    float* __restrict__ ws)           // [gridDim.x][128]: 64 colsum + 64 cnt
{
    __shared__ float lds[TOK_PER_BLOCK * LDS_PITCH];   // logits (+bias)
    __shared__ float row_m[TOK_PER_BLOCK];             // per-token max
    __shared__ float row_is[TOK_PER_BLOCK];            // per-token 1/sum(exp)
    __shared__ int   lds_cnt[N_EXP];                   // per-block expert counts

    const int tid  = threadIdx.x;
    const int wv   = tid >> 5;         // wave id 0..7
    const int lane = tid & 31;
    const int half = lane >> 4;        // 0: lanes 0-15, 1: lanes 16-31
    const int col  = lane & 15;
    const int tok0 = blockIdx.x * TOK_PER_BLOCK + wv * 16;

    if (tid < N_EXP) lds_cnt[tid] = 0;

    // ---------------- Phase A: 16-token x 64-expert GEMM via WMMA f32 -------
    // A 16x4 layout: lane<16 holds (k0,k1) of row M=col ; lane>=16 holds (k2,k3)
    // B  4x16 layout: V0 = rows {k0 | k0+2}, V1 = {k0+1 | k0+3}, N = col
    v8f acc0 = {}, acc1 = {}, acc2 = {}, acc3 = {};

    const float* aptr  = x   + (size_t)(tok0 + col) * H_DIM + half * 2;
    const float* bptr0 = wgt + (size_t)( 0 + col) * H_DIM + half * 2;
    const float* bptr1 = wgt + (size_t)(16 + col) * H_DIM + half * 2;
    const float* bptr2 = wgt + (size_t)(32 + col) * H_DIM + half * 2;
    const float* bptr3 = wgt + (size_t)(48 + col) * H_DIM + half * 2;

#pragma unroll 4
    for (int k = 0; k < H_DIM; k += 4) {
        v2f a  = *(const v2f*)(aptr  + k);
        v2f b0 = *(const v2f*)(bptr0 + k);
        v2f b1 = *(const v2f*)(bptr1 + k);
        v2f b2 = *(const v2f*)(bptr2 + k);
        v2f b3 = *(const v2f*)(bptr3 + k);
        acc0 = __builtin_amdgcn_wmma_f32_16x16x4_f32(false, a, false, b0, (short)0, acc0, false, false);
        acc1 = __builtin_amdgcn_wmma_f32_16x16x4_f32(false, a, false, b1, (short)0, acc1, false, false);
        acc2 = __builtin_amdgcn_wmma_f32_16x16x4_f32(false, a, false, b2, (short)0, acc2, false, false);
        acc3 = __builtin_amdgcn_wmma_f32_16x16x4_f32(false, a, false, b3, (short)0, acc3, false, false);
    }

    // C/D layout: vgpr r, lane -> token row = r + 8*half, expert = 16*t + col
    {
        const int rowBase = wv * 16 + half * 8;
        float bv0 = bias[ 0 + col], bv1 = bias[16 + col];
        float bv2 = bias[32 + col], bv3 = bias[48 + col];
#pragma unroll
        for (int r = 0; r < 8; ++r) {
            float* dst = &lds[(rowBase + r) * LDS_PITCH];
            dst[ 0 + col] = acc0[r] + bv0;
            dst[16 + col] = acc1[r] + bv1;
            dst[32 + col] = acc2[r] + bv2;
            dst[48 + col] = acc3[r] + bv3;
        }
    }
    __syncthreads();

    // ---------------- Phase B: softmax + top-8 (one thread per token) -------
    if (tid < TOK_PER_BLOCK) {
        const float* row = &lds[tid * LDS_PITCH];
        float mx = -3.402823e38f;
#pragma unroll 8
        for (int e = 0; e < N_EXP; ++e) mx = fmaxf(mx, row[e]);
        float s = 0.f;
#pragma unroll 8
        for (int e = 0; e < N_EXP; ++e) s += __expf(row[e] - mx);
        const float invs = 1.f / s;
        row_m[tid]  = mx;
        row_is[tid] = invs;

        unsigned long long mask = 0ull;
        int   idxs[TOPK];
        float scs[TOPK];
        float denom = 0.f;
#pragma unroll
        for (int i = 0; i < TOPK; ++i) {
            float best = -3.402823e38f; int bi = 0;
            for (int e = 0; e < N_EXP; ++e) {
                float v = row[e];
                bool taken = (mask >> e) & 1ull;
                if (!taken && v > best) { best = v; bi = e; }
            }
            mask |= (1ull << bi);
            idxs[i] = bi;
            scs[i]  = __expf(best - mx) * invs;
            denom  += scs[i];
        }
        const float dinv = 1.f / (denom + 1e-20f);
        const long long n = (long long)blockIdx.x * TOK_PER_BLOCK + tid;
#pragma unroll
        for (int i = 0; i < TOPK; ++i) {
            out_idx[n * TOPK + i] = idxs[i];
            out_w  [n * TOPK + i] = scs[i] * dinv;
            atomicAdd(&lds_cnt[idxs[i]], 1);     // LDS int atomic: exact
        }
    }
    __syncthreads();

    // ---------------- Phase C: per-block aux partials (deterministic) -------
    if (tid < N_EXP) {
        float colsum = 0.f;
        for (int t = 0; t < TOK_PER_BLOCK; ++t)
            colsum += __expf(lds[t * LDS_PITCH + tid] - row_m[t]) * row_is[t];
        float* slot = ws + (size_t)blockIdx.x * 128;
        slot[tid]          = colsum;              // sum of scores, this block
        slot[N_EXP + tid]  = (float)lds_cnt[tid]; // top-k hit count, this block
    }
}

// ---------------------------------------------------------------------------
// Deterministic aux-loss reduction: 256 threads = 4 batches x 64 experts
// ---------------------------------------------------------------------------
__global__ __launch_bounds__(256) void gate_aux_kernel(
    const float* __restrict__ ws, float* __restrict__ aux_out,
    int blocksPerBatch, int seqLen)
{
    __shared__ float red[256];
    const int t = threadIdx.x;
    const int b = t >> 6, e = t & 63;
    float sum = 0.f, cnt = 0.f;
    for (int blk = 0; blk < blocksPerBatch; ++blk) {
        const float* p = ws + (size_t)(b * blocksPerBatch + blk) * 128;
        sum += p[e];
        cnt += p[N_EXP + e];
    }
    // ce = cnt * E/(seq*K) ; contribution = ce * (sum/seq)
    float ce  = cnt * ((float)N_EXP / ((float)seqLen * (float)TOPK));
    float val = ce * (sum / (float)seqLen);
    red[t] = val;
    __syncthreads();
    for (int s = 128; s > 0; s >>= 1) {
        if (t < s) red[t] += red[t + s];
        __syncthreads();
    }
    if (t == 0) aux_out[0] = red[0] * 0.25f;   // mean over bsz=4
}

// ---------------------------------------------------------------------------
extern "C" void kernel_launch(void* const* d_in, const int* in_sizes, int n_in,
                              void* d_out, int out_size, void* d_ws, size_t ws_size,
                              hipStream_t stream) {
    const float* x    = (const float*)d_in[0];   // [4,4096,2048] f32
    const float* wgt  = (const float*)d_in[1];   // [64,2048] f32
    const float* bias = (const float*)d_in[2];   // [64] f32

    const int N = in_sizes[0] / H_DIM;           // 16384 tokens
    const int nBlocks = N / TOK_PER_BLOCK;       // 128
    const int bsz = 4;
    const int seqLen = N / bsz;                  // 4096

    int*   out_idx = (int*)d_out;                        // [N,8] int32
    float* out_w   = (float*)d_out + (size_t)N * TOPK;   // [N,8] f32
    float* out_aux = (float*)d_out + (size_t)2 * N * TOPK; // scalar
    float* ws      = (float*)d_ws;               // 128 blocks x 128 floats = 64 KB

    gate_main_kernel<<<nBlocks, 256, 0, stream>>>(x, wgt, bias, out_idx, out_w, ws);
    gate_aux_kernel<<<1, 256, 0, stream>>>(ws, out_aux, nBlocks / bsz, seqLen);
}